// GraphSAGE_87024627351878
// MI455X (gfx1250) — compile-verified
//
#include <hip/hip_runtime.h>

#define N_NODES 100000
#define FEATS   64
#define NEG_SLOPE 0.01f
#define WSTR    160   // dwords per k-pair row in LDS (128 data + 32 pad -> hi half-wave hits banks 32..63)

typedef __attribute__((ext_vector_type(2))) float v2f;
typedef __attribute__((ext_vector_type(8))) float v8f;

// ---------------------------------------------------------------------------
// zero a float buffer
// ---------------------------------------------------------------------------
__global__ __launch_bounds__(256) void zero_f32(float* __restrict__ p, int n) {
    int i = blockIdx.x * 256 + threadIdx.x;
    if (i < n) p[i] = 0.0f;
}

// ---------------------------------------------------------------------------
// in-degree (as float, matching segment_sum of ones)
// ---------------------------------------------------------------------------
__global__ __launch_bounds__(256) void sage_degree(const int* __restrict__ dst,
                                                   float* __restrict__ deg, int nEdges) {
    int e = blockIdx.x * 256 + threadIdx.x;
    if (e < nEdges) atomicAdd(&deg[dst[e]], 1.0f);
}

// ---------------------------------------------------------------------------
// edge aggregation: one wave32 per edge; coalesced 256B gather of h[src],
// 64 f32 atomic adds into neigh_sum[dst] (both L2-resident: 25.6MB each).
// ---------------------------------------------------------------------------
__global__ __launch_bounds__(256) void sage_aggregate(const float* __restrict__ h,
                                                      const int* __restrict__ src,
                                                      const int* __restrict__ dst,
                                                      float* __restrict__ ns, int nEdges) {
    int gwave = (blockIdx.x * 256 + threadIdx.x) >> 5;   // one wave per edge
    int lane  = threadIdx.x & 31;
    if (gwave >= nEdges) return;
    int s = src[gwave];
    int d = dst[gwave];
    const float2* hp = (const float2*)(h + (long long)s * FEATS);
    float2 v = hp[lane];
    float* outp = ns + (long long)d * FEATS + lane * 2;
    atomicAdd(outp + 0, v.x);
    atomicAdd(outp + 1, v.y);
}

// ---------------------------------------------------------------------------
// fused dual-GEMM + bias + leaky-relu:
//   out = h @ Wself^T + (ns/deg) @ Wneigh^T + b
// One wave per 16-node tile; 4 accumulator tiles span the 64 output features.
// fp32 WMMA (V_WMMA_F32_16X16X4_F32) -> exact f32 math; 128 wmma per wave.
// LDS weight layout is k-pair interleaved so each B fragment is a single
// aligned ds_load_b64 into an even VGPR pair (no repack movs).
// ---------------------------------------------------------------------------
__global__ __launch_bounds__(256) void sage_gemm(const float* __restrict__ h,
                                                 const float* __restrict__ ns,
                                                 const float* __restrict__ deg,
                                                 const float* __restrict__ wSelf,
                                                 const float* __restrict__ wNeigh,
                                                 const float* __restrict__ bias,
                                                 float* __restrict__ out,
                                                 int applyLeaky) {
    // lds[(k>>1)*WSTR + j*2 + (k&1)] = w[j*64 + k]
    __shared__ float ldsS[32 * WSTR];
    __shared__ float ldsN[32 * WSTR];
    for (int i = threadIdx.x; i < 64 * 64; i += 256) {
        int j = i >> 6, k = i & 63;
        int a = (k >> 1) * WSTR + j * 2 + (k & 1);
        ldsS[a] = wSelf[i];
        ldsN[a] = wNeigh[i];
    }
    __syncthreads();

    int wave  = threadIdx.x >> 5;
    int lane  = threadIdx.x & 31;
    int tile  = blockIdx.x * 8 + wave;    // 16-node row tile
    int node0 = tile * 16;
    if (node0 >= N_NODES) return;         // wave-uniform: EXEC stays all-ones

    int ln  = lane & 15;
    int kk  = (lane < 16) ? 0 : 2;        // A/B frag: lo lanes K={0,1}, hi lanes K={2,3}
    int row = node0 + ln;
    float inv = 1.0f / fmaxf(deg[row], 1.0f);   // exact IEEE divide (matches reference)

    const float* hrow = h  + (long long)row * FEATS + kk;
    const float* nrow = ns + (long long)row * FEATS + kk;

    v8f acc[4] = {};                      // zero accumulators (bias folded at store)

    for (int k0 = 0; k0 < FEATS; k0 += 4) {
        v2f a_s = *(const v2f*)(hrow + k0);        // global_load_b64, 8B aligned
        v2f a_n = *(const v2f*)(nrow + k0);
        a_n.x *= inv;
        a_n.y *= inv;
        int pbase = ((k0 + kk) >> 1) * WSTR + ln * 2;
#pragma unroll
        for (int j = 0; j < 4; ++j) {
            int pa = pbase + j * 32;               // (j0+ln)*2, j0 = 16j
            v2f bs = *(const v2f*)(ldsS + pa);     // ds_load_b64, even VGPR pair
            v2f bn = *(const v2f*)(ldsN + pa);
            acc[j] = __builtin_amdgcn_wmma_f32_16x16x4_f32(
                false, a_s, false, bs, (short)0, acc[j], false, false);
            acc[j] = __builtin_amdgcn_wmma_f32_16x16x4_f32(
                false, a_n, false, bn, (short)0, acc[j], false, false);
        }
    }

    // C layout: VGPR v -> M = v (lanes 0-15) / v+8 (lanes 16-31), N = ln
    int rbase = node0 + ((lane < 16) ? 0 : 8);
#pragma unroll
    for (int j = 0; j < 4; ++j) {
        int col = j * 16 + ln;
        float bb = bias[col];
#pragma unroll
        for (int v = 0; v < 8; ++v) {
            float val = acc[j][v] + bb;
            if (applyLeaky) val = (val >= 0.0f) ? val : NEG_SLOPE * val;
            out[(long long)(rbase + v) * FEATS + col] = val;
        }
    }
}

// ---------------------------------------------------------------------------
extern "C" void kernel_launch(void* const* d_in, const int* in_sizes, int n_in,
                              void* d_out, int out_size, void* d_ws, size_t ws_size,
                              hipStream_t stream) {
    const float* in_feat  = (const float*)d_in[0];
    const int*   src      = (const int*)d_in[1];
    const int*   dst      = (const int*)d_in[2];
    const float* w_self1  = (const float*)d_in[3];
    const float* w_neigh1 = (const float*)d_in[4];
    const float* b1       = (const float*)d_in[5];
    const float* w_self2  = (const float*)d_in[6];
    const float* w_neigh2 = (const float*)d_in[7];
    const float* b2       = (const float*)d_in[8];
    const float* w_self3  = (const float*)d_in[9];
    const float* w_neigh3 = (const float*)d_in[10];
    const float* b3       = (const float*)d_in[11];

    const int NE = in_sizes[1];           // 1,600,000 edges
    const int NF = N_NODES * FEATS;       // 6.4M floats

    float* ws  = (float*)d_ws;
    float* ns  = ws;                      // [N_NODES*64] neighbor sums
    float* deg = ns + NF;                 // [N_NODES]
    float* h1  = deg + N_NODES;           // [N_NODES*64]
    float* h2  = h1 + NF;                 // [N_NODES*64]

    const int gemmBlocks = (N_NODES / 16 + 7) / 8;         // 8 waves/block, 1 tile/wave
    const int aggBlocks  = (NE * 32 + 255) / 256;          // 1 wave/edge

    // degree (shared by all 3 layers)
    zero_f32<<<(N_NODES + 255) / 256, 256, 0, stream>>>(deg, N_NODES);
    sage_degree<<<(NE + 255) / 256, 256, 0, stream>>>(dst, deg, NE);

    // layer 1
    zero_f32<<<(NF + 255) / 256, 256, 0, stream>>>(ns, NF);
    sage_aggregate<<<aggBlocks, 256, 0, stream>>>(in_feat, src, dst, ns, NE);
    sage_gemm<<<gemmBlocks, 256, 0, stream>>>(in_feat, ns, deg, w_self1, w_neigh1, b1, h1, 1);

    // layer 2
    zero_f32<<<(NF + 255) / 256, 256, 0, stream>>>(ns, NF);
    sage_aggregate<<<aggBlocks, 256, 0, stream>>>(h1, src, dst, ns, NE);
    sage_gemm<<<gemmBlocks, 256, 0, stream>>>(h1, ns, deg, w_self2, w_neigh2, b2, h2, 1);

    // layer 3 (no activation)
    zero_f32<<<(NF + 255) / 256, 256, 0, stream>>>(ns, NF);
    sage_aggregate<<<aggBlocks, 256, 0, stream>>>(h2, src, dst, ns, NE);
    sage_gemm<<<gemmBlocks, 256, 0, stream>>>(h2, ns, deg, w_self3, w_neigh3, b3,
                                              (float*)d_out, 0);
}